// BackEdgeConv2d_74723841016060
// MI455X (gfx1250) — compile-verified
//
#include <hip/hip_runtime.h>
#include <cstdint>

// BackEdgeConv2d: threshold -> reflect-pad(3) -> 7x7 box sum -> band mask -> zero.
// Memory-bound (402 MB traffic, ~17us floor @ 23.3 TB/s). CDNA5 data-movement
// paths: TDM tensor_load_to_lds for interior tiles (rectangular 70x70 window),
// per-lane global_load_async_to_lds_b32 for reflect-border tiles.

#define IMG_W 1024
#define IMG_H 1024
#define TW 64
#define TH 64
#define HALO 3
#define SW (TW + 2*HALO)      // 70: staged tile width
#define SH (TH + 2*HALO)      // 70: staged tile height
#define NSTAGE (SW * SH)      // 4900 floats
#define RSW 65                // row-sum LDS stride (padded: kills bank conflicts)
#define THREADS 256

#if defined(__has_builtin)
#if __has_builtin(__builtin_amdgcn_tensor_load_to_lds) && __has_builtin(__builtin_amdgcn_s_wait_tensorcnt)
#define HAVE_TDM 1
#endif
#endif

#ifdef HAVE_TDM
typedef unsigned int u32x4 __attribute__((ext_vector_type(4)));
typedef int          i32x8 __attribute__((ext_vector_type(8)));
typedef int          i32x4 __attribute__((ext_vector_type(4)));
#endif

__device__ __forceinline__ int reflect_idx(int i, int n) {
    // jnp.pad mode='reflect' (edge not repeated): -1 -> 1, n -> n-2
    if (i < 0) i = -i;
    else if (i >= n) i = 2 * n - 2 - i;
    return i;
}

__global__ __launch_bounds__(THREADS)
void backedge_conv2d_kernel(const float* __restrict__ x, float* __restrict__ out) {
    __shared__ float stage[NSTAGE];     // raw x tile incl. halo (19600 B)
    __shared__ float rs[SH * RSW];      // horizontal 7-tap binary sums (18200 B)

    const int t   = threadIdx.x;
    const int tx0 = blockIdx.x * TW;
    const int ty0 = blockIdx.y * TH;
    const size_t plane = (size_t)blockIdx.z * (size_t)(IMG_H * IMG_W);
    const float* __restrict__ xp = x + plane;

    // ---- Phase 1: stage (TH+6)x(TW+6) tile (reflect-padded) into LDS ----
    const bool interior = (tx0 >= HALO) && (tx0 + TW + HALO <= IMG_W) &&
                          (ty0 >= HALO) && (ty0 + TH + HALO <= IMG_H);
#ifdef HAVE_TDM
    if (interior) {
        // TDM: one DMA descriptor moves the whole rectangular 70x70 window.
        // Wave 0 only (TDM ignores EXEC; gate with a *scalar* branch).
        if (__builtin_amdgcn_readfirstlane((int)t) == 0) {
            const uint64_t ga =
                (uint64_t)(uintptr_t)(xp + ((size_t)(ty0 - HALO) * IMG_W + (tx0 - HALO)));
            const uint32_t lds = (uint32_t)(uintptr_t)stage;
            // D# group 0: count=1 | lds_addr | global_addr[56:0] | type=2
            u32x4 g0 = { 1u, lds, (uint32_t)ga,
                         ((uint32_t)(ga >> 32) & 0x01FFFFFFu) | 0x80000000u };
            // D# group 1: data_size=2 (4B); tensor_dim0=1024, tensor_dim1=1024;
            // tile_dim0=70, tile_dim1=70, tile_dim2=0; tensor_dim0_stride=1024.
            i32x8 g1 = { (int)(2u << 16),                 // wg_mask=0, data_size=4B
                         (int)((1024u & 0xFFFFu) << 16),  // abar=0 | dim0.lo
                         (int)(((1024u >> 16) & 0xFFFFu) | ((1024u & 0xFFFFu) << 16)), // dim0.hi|dim1.lo
                         (int)(((1024u >> 16) & 0xFFFFu) | (70u << 16)),               // dim1.hi|tile0
                         (int)70u,                        // tile1 | tile2=0
                         (int)1024u,                      // dim0_stride lo
                         0, 0 };                          // stride hi, dim1_stride=0
            i32x4 gz4 = { 0, 0, 0, 0 };
            i32x8 gz8 = { 0, 0, 0, 0, 0, 0, 0, 0 };
            // clang-23 / therock lane: 6-arg form (g0, g1, g2, g3, extra, cpol)
            __builtin_amdgcn_tensor_load_to_lds(g0, g1, gz4, gz4, gz8, 0);
            __builtin_amdgcn_s_wait_tensorcnt(0);
            asm volatile("" ::: "memory");
        }
    } else
#endif
    {
        // Async per-lane copy handles reflect indexing at image borders
        // (and all tiles when the TDM builtin is unavailable).
        for (int i = t; i < NSTAGE; i += THREADS) {
            int ly = i / SW;
            int lx = i - ly * SW;
            int gy = interior ? (ty0 + ly - HALO) : reflect_idx(ty0 + ly - HALO, IMG_H);
            int gx = interior ? (tx0 + lx - HALO) : reflect_idx(tx0 + lx - HALO, IMG_W);
            const float* gptr = xp + (gy * IMG_W + gx);
            uint32_t lds_off = (uint32_t)(uintptr_t)(&stage[i]);
            asm volatile("global_load_async_to_lds_b32 %0, %1, off"
                         :: "v"(lds_off), "v"(gptr)
                         : "memory");
        }
        asm volatile("s_wait_asynccnt 0x0" ::: "memory");
    }
    __syncthreads();

    const float THRV = 128.0f / 255.0f;

    // ---- Phase 2: horizontal binarized 7-tap, sliding window over 16-col runs ----
    // 70 rows x 4 segments of 16 cols = 280 runs over 256 threads.
    for (int u = t; u < SH * 4; u += THREADS) {
        int row = u >> 2;
        int c0  = (u & 3) * 16;
        const float* sp = &stage[row * SW + c0];
        float s = 0.0f;
        #pragma unroll
        for (int d = 0; d < 7; ++d)
            s += (sp[d] >= THRV) ? 1.0f : 0.0f;
        float* rp = &rs[row * RSW + c0];
        rp[0] = s;
        #pragma unroll
        for (int c = 1; c < 16; ++c) {
            s += ((sp[c + 6] >= THRV) ? 1.0f : 0.0f)
               - ((sp[c - 1] >= THRV) ? 1.0f : 0.0f);
            rp[c] = s;
        }
    }
    __syncthreads();

    // ---- Phase 3: vertical 7-tap sliding window, mask, store ----
    // thread -> (column c, 16 contiguous rows starting at r0)
    const int c  = t & 63;
    const int r0 = (t >> 6) * 16;
    float cs = 0.0f;
    #pragma unroll
    for (int d = 0; d < 7; ++d)
        cs += rs[(r0 + d) * RSW + c];

    float* op = out + plane + (size_t)ty0 * IMG_W + tx0 + c;
    #pragma unroll
    for (int k = 0; k < 16; ++k) {
        int r = r0 + k;
        float xc = stage[(r + HALO) * SW + (c + HALO)];
        bool masked = (cs >= 4.8f) && (cs <= 19.2f);   // csum integer-valued: exact
        op[(size_t)r * IMG_W] = masked ? 0.0f : xc;
        if (k < 15)
            cs += rs[(r + 7) * RSW + c] - rs[r * RSW + c];
    }
}

extern "C" void kernel_launch(void* const* d_in, const int* in_sizes, int n_in,
                              void* d_out, int out_size, void* d_ws, size_t ws_size,
                              hipStream_t stream) {
    (void)n_in; (void)d_ws; (void)ws_size; (void)out_size;
    const float* x = (const float*)d_in[0];
    float* out = (float*)d_out;
    const int planes = in_sizes[0] / (IMG_H * IMG_W);   // B*C = 48
    dim3 grid(IMG_W / TW, IMG_H / TH, planes);
    dim3 block(THREADS);
    backedge_conv2d_kernel<<<grid, block, 0, stream>>>(x, out);
}